// DifferentiableLogicLayer_12111807775264
// MI455X (gfx1250) — compile-verified
//
#include <hip/hip_runtime.h>
#include <hip/hip_bf16.h>
#include <stdint.h>

#define B_DIM    2048
#define IN_DIM   8192
#define OUT_DIM  16384
#define N_GATES  16
#define ROWS     4          // batch rows per block; 4*8192*4B = 128 KB LDS -> 2 WGs/WGP
#define THREADS  256        // 8 waves (wave32)

// ---------------------------------------------------------------------------
// Kernel 1: per-column softmax + collapse 16 gate weights into 4 coefficients
//   out[i,j] = w_ab*a*b + w_a*a + w_b*b + w_c
// ---------------------------------------------------------------------------
__global__ void dll_coeff_kernel(const float* __restrict__ logits,
                                 float4* __restrict__ w) {
    const int j = blockIdx.x * blockDim.x + threadIdx.x;
    if (j >= OUT_DIM) return;
    const float* lp = logits + (size_t)j * N_GATES;
    float l[N_GATES];
    float m = -INFINITY;
#pragma unroll
    for (int k = 0; k < N_GATES; ++k) { l[k] = lp[k]; m = fmaxf(m, l[k]); }
    float s = 0.f;
#pragma unroll
    for (int k = 0; k < N_GATES; ++k) { l[k] = expf(l[k] - m); s += l[k]; }
    const float inv = 1.0f / s;
#pragma unroll
    for (int k = 0; k < N_GATES; ++k) l[k] *= inv;

    const float wab = l[1] - l[2] - l[4] - 2.f*l[6] - l[7] + l[8] + 2.f*l[9]
                    + l[11] + l[13] - l[14];
    const float wa  = l[2] + l[3] + l[6] + l[7] - l[8] - l[9] - l[12] - l[13];
    const float wb  = l[4] + l[5] + l[6] + l[7] - l[8] - l[9] - l[10] - l[11];
    const float wc  = l[8] + l[9] + l[10] + l[11] + l[12] + l[13] + l[14] + l[15];
    w[j] = make_float4(wab, wa, wb, wc);
}

// ---------------------------------------------------------------------------
// Kernel 2: stage 4 rows of x into LDS via gfx1250 async global->LDS DMA,
// then LDS-gather a/b per output column and write coalesced NT stores.
// ---------------------------------------------------------------------------
__global__ void __launch_bounds__(THREADS)
dll_main_kernel(const float* __restrict__ x,
                const float4* __restrict__ w,
                const int* __restrict__ a_idx,
                const int* __restrict__ b_idx,
                float* __restrict__ out) {
    extern __shared__ float tile[];            // ROWS * IN_DIM floats (128 KB)
    const int tid  = threadIdx.x;
    const int row0 = blockIdx.x * ROWS;

    // ---- Phase 1: async copy of ROWS contiguous rows (coalesced b128) ----
    // ROWS*IN_DIM*4B / 16B = 8192 chunks; 32 chunks per thread, stride-256.
    const float* src = x + (size_t)row0 * IN_DIM;
    for (int c = tid; c < (ROWS * IN_DIM) / 4; c += THREADS) {
        unsigned lds_addr = (unsigned)(uintptr_t)(tile + c * 4); // low 32b = LDS offset
        const float* gaddr = src + c * 4;
        asm volatile("global_load_async_to_lds_b128 %0, %1, off"
                     :: "v"(lds_addr), "v"(gaddr)
                     : "memory");
    }
    asm volatile("s_wait_asynccnt 0" ::: "memory");
    __syncthreads();

    // ---- Phase 2: gather + 4-term affine, all columns strided by THREADS --
    for (int j = tid; j < OUT_DIM; j += THREADS) {
        const float4 ww = w[j];                 // coalesced, L2-resident
        const int ai = a_idx[j];
        const int bi = b_idx[j];
        const float* ta = tile + ai;            // ds_load gathers
        const float* tb = tile + bi;
#pragma unroll
        for (int r = 0; r < ROWS; ++r) {
            const float a = ta[r * IN_DIM];
            const float b = tb[r * IN_DIM];
            const float v = fmaf(ww.x, a * b, fmaf(ww.y, a, fmaf(ww.z, b, ww.w)));
            // write-once output: non-temporal so x/idx/coeff stay hot in L2
            __builtin_nontemporal_store(v, &out[(size_t)(row0 + r) * OUT_DIM + j]);
        }
    }
}

// ---------------------------------------------------------------------------
extern "C" void kernel_launch(void* const* d_in, const int* in_sizes, int n_in,
                              void* d_out, int out_size, void* d_ws, size_t ws_size,
                              hipStream_t stream) {
    const float* x      = (const float*)d_in[0];
    const float* logits = (const float*)d_in[1];
    const int*   a_idx  = (const int*)d_in[2];
    const int*   b_idx  = (const int*)d_in[3];
    float*       out    = (float*)d_out;
    float4*      wcoef  = (float4*)d_ws;       // needs OUT_DIM*16 B = 256 KB

    dll_coeff_kernel<<<OUT_DIM / THREADS, THREADS, 0, stream>>>(logits, wcoef);

    const size_t lds_bytes = (size_t)ROWS * IN_DIM * sizeof(float); // 128 KB
    dll_main_kernel<<<B_DIM / ROWS, THREADS, lds_bytes, stream>>>(
        x, wcoef, a_idx, b_idx, out);
}